// random_masking_83786222010425
// MI455X (gfx1250) — compile-verified
//
#include <hip/hip_runtime.h>
#include <hip/hip_bf16.h>
#include <stdint.h>

// ---------------------------------------------------------------------------
// random_masking for (4,100,512,512) f32.
// Channels {0,10,...,90} are replaced with Gaussian noise, rest pass through.
// Pure bandwidth problem: ~797 MB HBM traffic -> ~34us floor at 23.3 TB/s.
// CDNA5 paths: async global<->LDS b128 DMA (ASYNCcnt), 4 transfers in flight
// per wave before each s_wait_asynccnt, NT cache policy on the whole stream;
// non-temporal b128 stores for the generated noise; wave32 blocks.
// ---------------------------------------------------------------------------

typedef float v4f __attribute__((ext_vector_type(4)));
typedef int   v4i __attribute__((vector_size(16)));   // matches builtin param

#define HW        262144u       // 512*512 floats per plane
#define Q_PLANE   65536u        // float4s per plane (HW/4)
#define NB        4u
#define NC        100u
#define NSEL      10u           // channels 0,10,...,90
#define NKEEP     90u
#define ILP       4u            // float4s per thread in the copy kernel
#define BLK       256u
#define CHUNK_Q   (ILP * BLK)   // 1024 float4s (16KB) per block

#define KEEP_Q    (NB * NKEEP * Q_PLANE)   // 23,592,960 float4s
#define NOISE_Q   (NB * NSEL  * Q_PLANE)   //  2,621,440 float4s

#define CPOL_NT   1             // TH[2:0] = NT (single-use stream, spare L2)

#if __has_builtin(__builtin_amdgcn_global_load_async_to_lds_b128) && \
    __has_builtin(__builtin_amdgcn_global_store_async_from_lds_b128)
#define USE_ASYNC_LDS 1
#pragma message("CDNA5 path: async global<->LDS b128 builtins AVAILABLE")
#else
#define USE_ASYNC_LDS 0
#pragma message("CDNA5 path: async LDS builtins MISSING -> nontemporal b128 copy")
#endif

__device__ __forceinline__ void wait_asynccnt0() {
#if __has_builtin(__builtin_amdgcn_s_wait_asynccnt)
    __builtin_amdgcn_s_wait_asynccnt(0);
#else
    asm volatile("s_wait_asynccnt 0x0" ::: "memory");
#endif
}

// --------------------------- pass-through copy -----------------------------
// Each block moves one contiguous 16KB chunk (1024 float4s) of one kept plane;
// 64 chunks per plane, 360 planes -> 23,040 blocks, no tail anywhere.
__global__ __launch_bounds__(BLK) void rm_copy_keep_kernel(
    const float* __restrict__ in, float* __restrict__ out) {
    const uint32_t tid   = threadIdx.x;
    const uint32_t chunk = blockIdx.x;            // 0..23039
    const uint32_t plane = chunk >> 6;            // 0..359 (64 chunks/plane)
    const uint32_t cw    = (chunk & 63u) << 10;   // chunk base within plane (f4)
    const uint32_t b     = plane / NKEEP;
    const uint32_t cc    = plane - b * NKEEP;     // 0..89
    const uint32_t c     = cc + cc / 9u + 1u;     // skip multiples of 10
    // base float4 index in the full tensor for this thread's first transfer
    const uint64_t q0 = (((uint64_t)(b * NC + c)) << 16) | (cw + tid);

#if USE_ASYNC_LDS
    // Wave-staged DMA copy: global -> LDS -> global, data never in VGPRs.
    // 4 coalesced async loads in flight per wave before one asynccnt wait.
    __shared__ __align__(16) char stage[CHUNK_Q * 16];

    __attribute__((address_space(1))) v4i* g_src =
        (__attribute__((address_space(1))) v4i*)(in) + q0;
    __attribute__((address_space(1))) v4i* g_dst =
        (__attribute__((address_space(1))) v4i*)(out) + q0;
    __attribute__((address_space(3))) v4i* l_ptr =
        (__attribute__((address_space(3))) v4i*)(stage) + tid;

#pragma unroll
    for (uint32_t j = 0; j < ILP; ++j)
        __builtin_amdgcn_global_load_async_to_lds_b128(
            g_src + j * BLK, l_ptr + j * BLK, 0, CPOL_NT);

    wait_asynccnt0();   // this wave's async loads complete in order

#pragma unroll
    for (uint32_t j = 0; j < ILP; ++j)
        __builtin_amdgcn_global_store_async_from_lds_b128(
            g_dst + j * BLK, l_ptr + j * BLK, 0, CPOL_NT);
    // store completion covered by implicit wait-idle at s_endpgm
#else
    const v4f* __restrict__ src = (const v4f*)in;
    v4f* __restrict__ dst       = (v4f*)out;
#pragma unroll
    for (uint32_t j = 0; j < ILP; ++j) {
        v4f v = __builtin_nontemporal_load(src + q0 + j * BLK);
        __builtin_nontemporal_store(v, dst + q0 + j * BLK);
    }
#endif
}

// ------------------------------ noise fill ---------------------------------
__device__ __forceinline__ uint32_t pcg_hash(uint32_t v) {
    uint32_t state = v * 747796405u + 2891336453u;
    uint32_t word  = ((state >> ((state >> 28u) + 4u)) ^ state) * 277803737u;
    return (word >> 22u) ^ word;
}
__device__ __forceinline__ float u01(uint32_t h) {
    // (0,1]: 24-bit mantissa, never zero -> __logf is finite
    return (float)((h >> 8) + 1u) * (1.0f / 16777216.0f);
}

__global__ __launch_bounds__(BLK) void rm_noise_fill_kernel(float* __restrict__ out) {
    const uint32_t i = blockIdx.x * BLK + threadIdx.x;   // noise-space float4 idx
    const uint32_t within = i & (Q_PLANE - 1u);
    const uint32_t plane  = i >> 16;                     // 0..39
    const uint32_t b      = plane / NSEL;
    const uint32_t k      = plane - b * NSEL;
    const uint32_t c      = 10u * k;                     // selected channel
    const uint64_t q = (((uint64_t)(b * NC + c)) << 16) | within;

    const uint32_t base = i * 4u;
    const float u0 = u01(pcg_hash(base + 0u));
    const float u1 = u01(pcg_hash(base + 1u));
    const float u2 = u01(pcg_hash(base + 2u));
    const float u3 = u01(pcg_hash(base + 3u));

    const float TWO_PI = 6.28318530717958647692f;
    float s0, c0, s1, c1;
    const float r0 = __fsqrt_rn(-2.0f * __logf(u0));
    __sincosf(TWO_PI * u1, &s0, &c0);
    const float r1 = __fsqrt_rn(-2.0f * __logf(u2));
    __sincosf(TWO_PI * u3, &s1, &c1);

    v4f z;
    z.x = r0 * c0;
    z.y = r0 * s0;
    z.z = r1 * c1;
    z.w = r1 * s1;
    __builtin_nontemporal_store(z, (v4f*)out + q);
}

// ------------------------------- launcher ----------------------------------
extern "C" void kernel_launch(void* const* d_in, const int* in_sizes, int n_in,
                              void* d_out, int out_size, void* d_ws, size_t ws_size,
                              hipStream_t stream) {
    (void)in_sizes; (void)n_in; (void)out_size; (void)d_ws; (void)ws_size;
    const float* in = (const float*)d_in[0];
    float* out      = (float*)d_out;

    // KEEP_Q / CHUNK_Q = 23,040 blocks ; NOISE_Q / BLK = 10,240 blocks (exact)
    rm_copy_keep_kernel<<<KEEP_Q / CHUNK_Q, BLK, 0, stream>>>(in, out);
    rm_noise_fill_kernel<<<NOISE_Q / BLK, BLK, 0, stream>>>(out);
}